// _MultiHeadAttention_58712202936484
// MI455X (gfx1250) — compile-verified
//
#include <hip/hip_runtime.h>

#define B_   2
#define S_   2048
#define H_   16
#define D_   64
#define R_   1024
#define NQKV 192

typedef float v2f __attribute__((ext_vector_type(2)));
typedef float v8f __attribute__((ext_vector_type(8)));
typedef unsigned u32x4 __attribute__((ext_vector_type(4)));
typedef int      i32x4 __attribute__((ext_vector_type(4)));
typedef int      i32x8 __attribute__((ext_vector_type(8)));

// Native fp32 WMMA: D(16x16 f32) = A(16x4 f32) * B(4x16 f32) + C
static __device__ __forceinline__ v8f wmma4(v2f a, v2f b, v8f c) {
  return __builtin_amdgcn_wmma_f32_16x16x4_f32(
      false, a, false, b, (short)0, c, false, false);
}

// Generic (flat) pointer to a __shared__ object: low 32 bits are the LDS
// offset (shared aperture lives in the high 32 bits on gfx1250).
static __device__ __forceinline__ unsigned lds_off(const void* p) {
  return (unsigned)(size_t)p;
}

// Async global->LDS scatter copy (4B per lane), tracked by ASYNCcnt.
static __device__ __forceinline__ void async_b32(unsigned lds, const float* g) {
  asm volatile("global_load_async_to_lds_b32 %0, %1, off"
               :: "v"(lds), "v"(g) : "memory");
}

// 2D tile load via the Tensor Data Mover (issue from one wave per WG).
//   tile_w floats per row, tile_h rows, global row stride in floats.
//   LDS padding: +(pad_amt_m1+1) dwords after every 2^(pad_log+1) dwords.
static __device__ __forceinline__ void tdm_load_2d(
    const float* lds_dst, const float* gsrc,
    unsigned tile_w, unsigned tile_h, unsigned stride,
    unsigned pad_log, unsigned pad_amt_m1)
{
  unsigned long long ga = (unsigned long long)gsrc;
  u32x4 g0;
  g0[0] = 1u;                                        // count=1 (valid user D#)
  g0[1] = lds_off(lds_dst);                          // lds_addr
  g0[2] = (unsigned)ga;                              // global_addr[31:0]
  g0[3] = (unsigned)((ga >> 32) & 0x01FFFFFFull)     // global_addr[56:32]
        | (2u << 30);                                // type=2 ("image")
  i32x8 g1;
  g1[0] = (int)((2u << 16)                           // data_size = 4B
              | (1u << 20)                           // pad_enable
              | (pad_log << 22)                      // pad_interval
              | (pad_amt_m1 << 25));                 // pad_amount
  g1[1] = (int)((tile_w & 0xFFFFu) << 16);           // tensor_dim0[15:0]
  g1[2] = (int)(((tile_w >> 16) & 0xFFFFu)           // tensor_dim0[31:16]
              | ((tile_h & 0xFFFFu) << 16));         // tensor_dim1[15:0]
  g1[3] = (int)(((tile_h >> 16) & 0xFFFFu)           // tensor_dim1[31:16]
              | ((tile_w & 0xFFFFu) << 16));         // tile_dim0
  g1[4] = (int)(tile_h & 0xFFFFu);                   // tile_dim1 (tile_dim2=0)
  g1[5] = (int)stride;                               // tensor_dim0_stride[31:0]
  g1[6] = 0;
  g1[7] = 0;
  i32x4 gz4 = {0, 0, 0, 0};
  i32x8 gz8 = {0, 0, 0, 0, 0, 0, 0, 0};
  // amdgpu-toolchain (clang-23) 6-arg form:
  //   (g0, g1, g2, g3, extra group, cpol)
  __builtin_amdgcn_tensor_load_to_lds(g0, g1, gz4, gz4, gz8, 0);
}

// ---------------------------------------------------------------------------
// Tiled GEMM: C = A(4096 x 1024) * Bmat(1024 x N) + bias
//   mode 0: QKV projection (per-head weight slice, qkv-layout store)
//   mode 1: output projection (plain row-major store)
// 256 threads = 8 waves; tile 128(M) x 64(N); K in 32-chunks, double-buffered:
//   A chunk via TDM (padded 36-stride), B chunk via async b32 transpose.
// ---------------------------------------------------------------------------
__global__ __launch_bounds__(256) void gemm_kernel(
    const float* __restrict__ A, const float* __restrict__ Bw,
    const float* __restrict__ bias, float* __restrict__ C,
    int ldb, int mode)
{
  __shared__ float As[2][128 * 36];
  __shared__ float Bs[2][64 * 36];

  const int tid  = threadIdx.x;
  const int lane = tid & 31;
  const int w    = tid >> 5;
  const int wm   = w & 3;
  const int wn   = w >> 2;
  const int l15  = lane & 15;
  const int kh   = (lane >> 4) << 1;

  const int mBase = blockIdx.x * 128;
  const int nBase = blockIdx.y * 64;
  const int head  = blockIdx.z;
  const float* Bmat = (mode == 0) ? (Bw + (size_t)head * R_ * NQKV) : Bw;

  // Prologue: stage chunk 0 into buffer 0.
  if (w == 0)
    tdm_load_2d(&As[0][0], A + (size_t)mBase * R_, 32, 128, R_, 4, 3);
  #pragma unroll
  for (int i = 0; i < 8; ++i) {
    int idx = tid + i * 256;
    int k = idx >> 6, n = idx & 63;
    async_b32(lds_off(&Bs[0][n * 36 + k]),
              Bmat + (size_t)k * ldb + nBase + n);
  }

  v8f acc00 = {}, acc01 = {}, acc10 = {}, acc11 = {};

  for (int kc = 0; kc < 32; ++kc) {
    const int cur = kc & 1;
    if (kc < 31) {
      const int k0n = (kc + 1) * 32;
      if (w == 0)
        tdm_load_2d(&As[1 - cur][0], A + (size_t)mBase * R_ + k0n,
                    32, 128, R_, 4, 3);
      #pragma unroll
      for (int i = 0; i < 8; ++i) {
        int idx = tid + i * 256;
        int k = idx >> 6, n = idx & 63;
        async_b32(lds_off(&Bs[1 - cur][n * 36 + k]),
                  Bmat + (size_t)(k0n + k) * ldb + nBase + n);
      }
      if (w == 0) __builtin_amdgcn_s_wait_tensorcnt(1);
      asm volatile("s_wait_asynccnt 0x8" ::: "memory");
    } else {
      if (w == 0) __builtin_amdgcn_s_wait_tensorcnt(0);
      asm volatile("s_wait_asynccnt 0x0" ::: "memory");
    }
    __syncthreads();                 // cur buffer fully staged, all waves

    const float* as = &As[cur][0];
    const float* bs = &Bs[cur][0];
    #pragma unroll
    for (int kk = 0; kk < 32; kk += 4) {
      v2f a0 = *(const v2f*)(as + (wm * 32      + l15) * 36 + kk + kh);
      v2f a1 = *(const v2f*)(as + (wm * 32 + 16 + l15) * 36 + kk + kh);
      v2f b0 = *(const v2f*)(bs + (wn * 32      + l15) * 36 + kk + kh);
      v2f b1 = *(const v2f*)(bs + (wn * 32 + 16 + l15) * 36 + kk + kh);
      acc00 = wmma4(a0, b0, acc00);
      acc01 = wmma4(a0, b1, acc01);
      acc10 = wmma4(a1, b0, acc10);
      acc11 = wmma4(a1, b1, acc11);
    }
    __syncthreads();                 // done with cur before it is re-staged
  }

  // Store. C layout: VGPR r -> M=r (lanes 0-15) / M=r+8 (lanes 16-31).
  const int mlHi = (lane >> 4) << 3;
  #pragma unroll
  for (int i = 0; i < 2; ++i) {
    #pragma unroll
    for (int j = 0; j < 2; ++j) {
      v8f acc = (i == 0) ? (j == 0 ? acc00 : acc01)
                         : (j == 0 ? acc10 : acc11);
      #pragma unroll
      for (int r = 0; r < 8; ++r) {
        int m = mBase + wm * 32 + i * 16 + r + mlHi;
        int n = nBase + wn * 32 + j * 16 + l15;
        if (mode == 0) {
          int b = m >> 11, s = m & (S_ - 1);
          C[(((size_t)(b * H_ + head)) * S_ + s) * NQKV + n]
              = acc[r] + bias[head * NQKV + n];
        } else {
          C[(size_t)m * R_ + n] = acc[r] + bias[n];
        }
      }
    }
  }
}

// ---------------------------------------------------------------------------
// Fused flash attention for one (b,h). 128 threads = 4 waves; 64 q rows per
// block (16/wave, Q held in registers in A-fragment layout); 64-key blocks.
// K double-buffered via TDM prefetch; V staged transposed via async b32;
// P reuses the current K buffer (its prefetch target is the other buffer).
// ---------------------------------------------------------------------------
__global__ __launch_bounds__(128) void attn_kernel(
    const float* __restrict__ qkv, float* __restrict__ out)
{
  __shared__ float KB[2][64 * 68];   // K block (double buffer); P reuses cur
  __shared__ float Vt[64 * 68];      // V transposed: Vt[d][key]

  const int tid  = threadIdx.x;
  const int lane = tid & 31;
  const int w    = tid >> 5;             // 0..3
  const int l15  = lane & 15;
  const int kh   = (lane >> 4) << 1;
  const int mlHi = (lane >> 4) << 3;

  const int bh = blockIdx.y;
  const int b  = bh >> 4, h = bh & 15;
  const int sBase = blockIdx.x * 64;
  const float* base = qkv + (size_t)bh * S_ * NQKV;

  // Q rows for this wave straight into registers (A-fragment layout):
  // lane holds Q[16w + l15][kd + kh .. kd + kh + 1] for kd = 0,4,...,60.
  v2f qreg[16];
  {
    const float* qrow = base + (size_t)(sBase + w * 16 + l15) * NQKV;
    #pragma unroll
    for (int t = 0; t < 16; ++t)
      qreg[t] = *(const v2f*)(qrow + t * 4 + kh);
  }

  // Prologue: TDM K block 0 -> KB[0] (padded to 68-stride).
  if (w == 0)
    tdm_load_2d(&KB[0][0], base + D_, 64, 64, NQKV, 5, 3);

  v8f o0 = {}, o1 = {}, o2 = {}, o3 = {};
  float mrun[8], lrun[8];
  #pragma unroll
  for (int r = 0; r < 8; ++r) { mrun[r] = -1e30f; lrun[r] = 0.0f; }

  for (int kb = 0; kb < S_; kb += 64) {
    const int cur = (kb >> 6) & 1;
    __syncthreads();                 // previous iteration's PV fully done

    // Prefetch next K block into the other buffer (holds consumed P).
    if (kb + 64 < S_) {
      if (w == 0)
        tdm_load_2d(&KB[1 - cur][0],
                    base + (size_t)(kb + 64) * NQKV + D_, 64, 64, NQKV, 5, 3);
    }
    // Async transpose-stage V for the *current* block; waited before PV,
    // so its latency hides behind scores + softmax.
    #pragma unroll
    for (int i = 0; i < 32; ++i) {
      int idx = tid + i * 128;
      int d = idx & 63, key = idx >> 6;
      async_b32(lds_off(&Vt[d * 68 + key]),
                base + (size_t)(kb + key) * NQKV + 2 * D_ + d);
    }
    if (w == 0) {
      if (kb + 64 < S_) __builtin_amdgcn_s_wait_tensorcnt(1);
      else              __builtin_amdgcn_s_wait_tensorcnt(0);
    }
    __syncthreads();                 // K[cur] visible to all waves

    // Scores: 16 q rows x 64 keys, reduce over head dim (K=4 per wmma).
    const float* ks = &KB[cur][0];
    v8f sc0 = {}, sc1 = {}, sc2 = {}, sc3 = {};
    #pragma unroll
    for (int kd = 0; kd < 64; kd += 4) {
      v2f a  = qreg[kd >> 2];
      v2f b0 = *(const v2f*)(ks + ( 0 + l15) * 68 + kd + kh);
      v2f b1 = *(const v2f*)(ks + (16 + l15) * 68 + kd + kh);
      v2f b2 = *(const v2f*)(ks + (32 + l15) * 68 + kd + kh);
      v2f b3 = *(const v2f*)(ks + (48 + l15) * 68 + kd + kh);
      sc0 = wmma4(a, b0, sc0);
      sc1 = wmma4(a, b1, sc1);
      sc2 = wmma4(a, b2, sc2);
      sc3 = wmma4(a, b3, sc3);
    }
    __syncthreads();                 // all K reads done; P may overwrite

    const float scale = 0.125f;      // 1/sqrt(64)
    float* ps = &KB[cur][0];
    #pragma unroll
    for (int r = 0; r < 8; ++r) {
      float s0 = sc0[r] * scale, s1 = sc1[r] * scale;
      float s2 = sc2[r] * scale, s3 = sc3[r] * scale;
      float mx = fmaxf(fmaxf(s0, s1), fmaxf(s2, s3));
      #pragma unroll
      for (int off = 8; off >= 1; off >>= 1)
        mx = fmaxf(mx, __shfl_xor(mx, off, 32));   // within 16-lane half
      float mnew = fmaxf(mrun[r], mx);
      float corr = __expf(mrun[r] - mnew);
      mrun[r] = mnew;
      float p0 = __expf(s0 - mnew), p1 = __expf(s1 - mnew);
      float p2 = __expf(s2 - mnew), p3 = __expf(s3 - mnew);
      float rs = p0 + p1 + p2 + p3;
      #pragma unroll
      for (int off = 8; off >= 1; off >>= 1)
        rs += __shfl_xor(rs, off, 32);
      lrun[r] = lrun[r] * corr + rs;
      o0[r] *= corr; o1[r] *= corr; o2[r] *= corr; o3[r] *= corr;
      int row = w * 16 + r + mlHi;               // C-layout row of this lane
      ps[row * 68 +  0 + l15] = p0;
      ps[row * 68 + 16 + l15] = p1;
      ps[row * 68 + 32 + l15] = p2;
      ps[row * 68 + 48 + l15] = p3;
    }
    __builtin_amdgcn_wave_barrier();
    asm volatile("s_wait_asynccnt 0x0" ::: "memory");  // this wave's Vt landed
    __syncthreads();                 // all waves' Vt stores visible

    // O += P(16x64) @ V(64x64): reduce over the 64 keys of this block.
    #pragma unroll
    for (int kss = 0; kss < 64; kss += 4) {
      v2f a  = *(const v2f*)(ps + (w * 16 + l15) * 68 + kss + kh);
      v2f b0 = *(const v2f*)(&Vt[( 0 + l15) * 68 + kss + kh]);
      v2f b1 = *(const v2f*)(&Vt[(16 + l15) * 68 + kss + kh]);
      v2f b2 = *(const v2f*)(&Vt[(32 + l15) * 68 + kss + kh]);
      v2f b3 = *(const v2f*)(&Vt[(48 + l15) * 68 + kss + kh]);
      o0 = wmma4(a, b0, o0);
      o1 = wmma4(a, b1, o1);
      o2 = wmma4(a, b2, o2);
      o3 = wmma4(a, b3, o3);
    }
  }

  // Normalize and store heads-concatenated: out[b][s][h*64 + d].
  #pragma unroll
  for (int r = 0; r < 8; ++r) {
    float inv = 1.0f / lrun[r];
    int s = sBase + w * 16 + r + mlHi;
    size_t o = ((size_t)b * S_ + s) * (H_ * D_) + (size_t)h * D_;
    out[o +  0 + l15] = o0[r] * inv;
    out[o + 16 + l15] = o1[r] * inv;
    out[o + 32 + l15] = o2[r] * inv;
    out[o + 48 + l15] = o3[r] * inv;
  }
}

// ---------------------------------------------------------------------------
extern "C" void kernel_launch(void* const* d_in, const int* in_sizes, int n_in,
                              void* d_out, int out_size, void* d_ws, size_t ws_size,
                              hipStream_t stream) {
  const float* x    = (const float*)d_in[0];   // (2, 2048, 1024)
  const float* Wqkv = (const float*)d_in[1];   // (16, 1024, 192)
  const float* bqkv = (const float*)d_in[2];   // (16, 192)
  const float* Wo   = (const float*)d_in[3];   // (1024, 1024)
  const float* bo   = (const float*)d_in[4];   // (1024,)
  float* out = (float*)d_out;                  // (2, 2048, 1024)

  float* qkvbuf  = (float*)d_ws;                                  // 48 MB
  float* attnbuf = qkvbuf + (size_t)B_ * H_ * S_ * NQKV;          // 16 MB

  // 1) QKV projection: per-head GEMM into (b,h,s,192) layout.
  dim3 g0(32, 3, 16);
  gemm_kernel<<<g0, 256, 0, stream>>>(x, Wqkv, bqkv, qkvbuf, NQKV, 0);

  // 2) Fused flash attention -> (b,s,1024).
  dim3 g1(32, B_ * H_, 1);
  attn_kernel<<<g1, 128, 0, stream>>>(qkvbuf, attnbuf);

  // 3) Output projection.
  dim3 g2(32, 16, 1);
  gemm_kernel<<<g2, 16 * 16, 0, stream>>>(attnbuf, Wo, bo, out, R_, 1);
}